// NewTripletLoss_56289841382063
// MI455X (gfx1250) — compile-verified
//
#include <hip/hip_runtime.h>

typedef __attribute__((ext_vector_type(2))) float v2f;
typedef __attribute__((ext_vector_type(8))) float v8f;

#define DIMS    128
#define KSTEPS  32          // 128 / 4 per V_WMMA_F32_16X16X4_F32
#define NSPLIT  8
#define TMARGIN 0.3f

// -------- prolog: per-row sum of squares + init mining state --------
__global__ __launch_bounds__(256) void triplet_prolog(
    const float* __restrict__ X, float* __restrict__ sq,
    unsigned* __restrict__ maxp, unsigned* __restrict__ minn, int n)
{
    const int wid  = (blockIdx.x * blockDim.x + threadIdx.x) >> 5;  // one wave per row
    const int lane = threadIdx.x & 31;
    if (wid >= n) return;
    const float4 v = *(const float4*)(X + (size_t)wid * DIMS + lane * 4);
    float s = v.x * v.x + v.y * v.y + v.z * v.z + v.w * v.w;
#pragma unroll
    for (int m = 16; m >= 1; m >>= 1) s += __shfl_xor(s, m, 32);
    if (lane == 0) {
        sq[wid]   = s;
        maxp[wid] = 0u;           // bits of +0.0f  (d2 >= 0)
        minn[wid] = 0x7F800000u;  // bits of +inf
    }
}

// -------- pack: swizzle X into WMMA panel layout Xp[tile][k][lane] --------
// Panel element (tile,k,lane) = float2 at X[tile*16 + lane%16][4k + 2*(lane/16)]
// == exactly what lane L must hold in VGPR-pair k for both A and B operands of
// V_WMMA_F32_16X16X4_F32. One gather per tile here, coalesced loads forever after.
__global__ __launch_bounds__(256) void triplet_pack(
    const float* __restrict__ X, v2f* __restrict__ Xp, int ntiles)
{
    const int wid  = (blockIdx.x * blockDim.x + threadIdx.x) >> 5;  // one wave per tile
    const int lane = threadIdx.x & 31;
    if (wid >= ntiles) return;
    const int half = lane >> 4, lmod = lane & 15;
    const float* row = X + (size_t)(wid * 16 + lmod) * DIMS + 2 * half;
    v2f* dst = Xp + (size_t)wid * (KSTEPS * 32) + lane;
#pragma unroll
    for (int k = 0; k < KSTEPS; ++k)
        dst[k * 32] = *(const v2f*)(row + 4 * k);   // 256B coalesced store
}

// -------- main: fused f32-WMMA Gram tiles + batch-hard mining --------
__global__ __launch_bounds__(256) void triplet_main(
    const v2f* __restrict__ Xp, const float* __restrict__ sq,
    const int* __restrict__ tgt, unsigned* __restrict__ maxp,
    unsigned* __restrict__ minn, int tiles_per_split)
{
    const int lane = threadIdx.x & 31;
    const int wid  = threadIdx.x >> 5;      // 8 waves / block, one 16-row tile each
    const int half = lane >> 4;
    const int lmod = lane & 15;

    const int m_tile = blockIdx.x * 8 + wid;
    const int m_base = m_tile * 16;

    // A panel: 32 coalesced b64 loads, lives in 64 VGPRs for the whole kernel
    const v2f* PA = Xp + (size_t)m_tile * (KSTEPS * 32) + lane;
    v2f a[KSTEPS];
#pragma unroll
    for (int k = 0; k < KSTEPS; ++k) a[k] = PA[k * 32];

    float sqm[8]; int tm[8];
#pragma unroll
    for (int r = 0; r < 8; ++r) {           // C-layout rows: M = r + 8*half
        const int row = m_base + r + 8 * half;
        sqm[r] = sq[row];
        tm[r]  = tgt[row];
    }

    float mp[8], mn[8];
#pragma unroll
    for (int r = 0; r < 8; ++r) { mp[r] = 0.0f; mn[r] = __builtin_inff(); }

    const int nt0 = blockIdx.y * tiles_per_split;
    const int nt1 = nt0 + tiles_per_split;   // tiles_per_split is even

    v2f b0[KSTEPS], b1[KSTEPS];
    {   // prime the pipeline
        const v2f* p = Xp + (size_t)nt0 * (KSTEPS * 32) + lane;
#pragma unroll
        for (int k = 0; k < KSTEPS; ++k) b0[k] = p[k * 32];
    }

    for (int nt = nt0; nt < nt1; nt += 2) {
        // ---- issue next tile's loads (b1) before consuming b0 ----
        const v2f* p1 = Xp + (size_t)(nt + 1) * (KSTEPS * 32) + lane;
        __builtin_prefetch(p1 + 2 * KSTEPS * 32, 0, 1);   // warm L0 two panels out
#pragma unroll
        for (int k = 0; k < KSTEPS; ++k) b1[k] = p1[k * 32];

        {   // ---- tile nt from b0: 32 back-to-back WMMAs ----
            v8f acc = {0.f, 0.f, 0.f, 0.f, 0.f, 0.f, 0.f, 0.f};
#pragma unroll
            for (int k = 0; k < KSTEPS; ++k)
                acc = __builtin_amdgcn_wmma_f32_16x16x4_f32(
                    false, a[k], false, b0[k], (short)0, acc, false, false);
            const int   ncol = nt * 16 + lmod;
            const float sqn  = sq[ncol];
            const int   tn   = tgt[ncol];
#pragma unroll
            for (int r = 0; r < 8; ++r) {
                float d2 = fmaf(-2.0f, acc[r], sqm[r] + sqn);
                d2 = fmaxf(d2, 0.0f);
                const bool same = (tm[r] == tn);
                mp[r] = fmaxf(mp[r], same ? d2 : 0.0f);
                mn[r] = fminf(mn[r], same ? __builtin_inff() : d2);
            }
        }

        // ---- refill b0 for tile nt+2 while b1's WMMAs run ----
        if (nt + 2 < nt1) {
            const v2f* p0 = Xp + (size_t)(nt + 2) * (KSTEPS * 32) + lane;
#pragma unroll
            for (int k = 0; k < KSTEPS; ++k) b0[k] = p0[k * 32];
        }

        {   // ---- tile nt+1 from b1 ----
            v8f acc = {0.f, 0.f, 0.f, 0.f, 0.f, 0.f, 0.f, 0.f};
#pragma unroll
            for (int k = 0; k < KSTEPS; ++k)
                acc = __builtin_amdgcn_wmma_f32_16x16x4_f32(
                    false, a[k], false, b1[k], (short)0, acc, false, false);
            const int   ncol = (nt + 1) * 16 + lmod;
            const float sqn  = sq[ncol];
            const int   tn   = tgt[ncol];
#pragma unroll
            for (int r = 0; r < 8; ++r) {
                float d2 = fmaf(-2.0f, acc[r], sqm[r] + sqn);
                d2 = fmaxf(d2, 0.0f);
                const bool same = (tm[r] == tn);
                mp[r] = fmaxf(mp[r], same ? d2 : 0.0f);
                mn[r] = fminf(mn[r], same ? __builtin_inff() : d2);
            }
        }
    }

    // reduce across the 16 lanes that share each row (masks 1,2,4,8)
#pragma unroll
    for (int s = 1; s <= 8; s <<= 1) {
#pragma unroll
        for (int r = 0; r < 8; ++r) {
            mp[r] = fmaxf(mp[r], __shfl_xor(mp[r], s, 32));
            mn[r] = fminf(mn[r], __shfl_xor(mn[r], s, 32));
        }
    }

    if (lmod == 0) {  // lanes 0 and 16 publish rows m_base+r and m_base+8+r
#pragma unroll
        for (int r = 0; r < 8; ++r) {
            const int row = m_base + r + 8 * half;
            atomicMax(&maxp[row], __float_as_uint(mp[r]));  // monotone for d2>=0
            atomicMin(&minn[row], __float_as_uint(mn[r]));
        }
    }
}

// -------- finalize: sqrt(clip) + margin relu + mean, one block --------
__global__ __launch_bounds__(1024) void triplet_finalize(
    const unsigned* __restrict__ maxp, const unsigned* __restrict__ minn,
    float* __restrict__ out, int n)
{
    __shared__ float smem[32];
    float acc = 0.0f;
    for (int i = threadIdx.x; i < n; i += 1024) {
        const float dap = sqrtf(fmaxf(__uint_as_float(maxp[i]), 1e-12f));
        const float dan = sqrtf(fmaxf(__uint_as_float(minn[i]), 1e-12f));
        acc += fmaxf(dap - dan + TMARGIN, 0.0f);
    }
#pragma unroll
    for (int m = 16; m >= 1; m >>= 1) acc += __shfl_xor(acc, m, 32);
    const int lane = threadIdx.x & 31, w = threadIdx.x >> 5;
    if (lane == 0) smem[w] = acc;
    __syncthreads();
    if (w == 0) {
        float a = smem[lane];
#pragma unroll
        for (int m = 16; m >= 1; m >>= 1) a += __shfl_xor(a, m, 32);
        if (lane == 0) out[0] = a / (float)n;
    }
}

extern "C" void kernel_launch(void* const* d_in, const int* in_sizes, int n_in,
                              void* d_out, int out_size, void* d_ws, size_t ws_size,
                              hipStream_t stream) {
    const float* X   = (const float*)d_in[0];   // [N, 128] fp32
    const int*   tgt = (const int*)d_in[1];     // [N] labels
    float*       out = (float*)d_out;           // scalar loss
    const int N = in_sizes[1];                  // 8192

    // workspace: sq[N] | maxp[N] | minn[N] | Xp[N*128] (panel-packed copy)
    float*    sq   = (float*)d_ws;
    unsigned* maxp = (unsigned*)((char*)d_ws + (size_t)N * 4);
    unsigned* minn = maxp + N;
    v2f*      Xp   = (v2f*)((char*)d_ws + (size_t)3 * N * 4);

    const int ntiles          = N / 16;               // 512
    const int tiles_per_split = ntiles / NSPLIT;      // 64 (even)

    triplet_prolog<<<N / 8, 256, 0, stream>>>(X, sq, maxp, minn, N);
    triplet_pack<<<(ntiles + 7) / 8, 256, 0, stream>>>(X, Xp, ntiles);
    triplet_main<<<dim3(N / 128, NSPLIT), 256, 0, stream>>>(
        Xp, sq, tgt, maxp, minn, tiles_per_split);
    triplet_finalize<<<1, 1024, 0, stream>>>(maxp, minn, out, N);
    (void)n_in; (void)out_size; (void)ws_size;
}